// YOLOv3_88124138979435
// MI455X (gfx1250) — compile-verified
//
#include <hip/hip_runtime.h>

// ---------------------------------------------------------------------------
// YOLOv3 loss for MI455X (gfx1250, wave32).
//  * prep kernel: per-(b,k) label preprocessing (anchor argmax, target vals),
//    packed into one 2048B per-batch blob for async LDS staging.
//  * main kernel: stages the blob via GLOBAL_LOAD_ASYNC_TO_LDS_B128
//    (ASYNCcnt-tracked DMA, no VGPR round trip), per-cell sigmoid/IoU/BCE
//    with lazy class-channel reads (only ~40 obj cells per image touch the
//    80 class channels -> ~94MB of HBM traffic avoided), and a
//    V_WMMA_F32_16X16X4_F32 ones-matrix wave reduction feeding atomics.
// ---------------------------------------------------------------------------

#define N_A         3
#define N_CLS       80
#define NCH         (5 + N_CLS)
#define K_LAB       50
#define NB          16
#define NG          76
#define IGNORE_THRE 0.6f
#define EPSV        1e-7f

// Per-batch metadata blob layout (bytes):
//   [0,800)     float4 gt[K_LAB]
//   [800,1600)  float4 tgt[K_LAB]
//   [1600,1800) float  tscale[K_LAB]
//   [1800,2000) int    meta[K_LAB]
//   pad to 2048 (128 x 16B async chunks)
#define BLOB_GT     0
#define BLOB_TGT    800
#define BLOB_TS     1600
#define BLOB_META   1800
#define BLOB_STRIDE 2048

typedef __attribute__((ext_vector_type(2))) float v2f;
typedef __attribute__((ext_vector_type(8))) float v8f;
typedef __attribute__((ext_vector_type(4))) int   v4i;

__device__ __forceinline__ float sigmoidf(float x) {
  return 1.0f / (1.0f + __expf(-x));
}

__device__ __forceinline__ float bcef(float p, float t) {
  p = fminf(fmaxf(p, EPSV), 1.0f - EPSV);
  return -(t * __logf(p) + (1.0f - t) * __logf(1.0f - p));
}

// ---------------------------------------------------------------------------
// Kernel 1: label preprocessing. One block per batch image, K_LAB labels.
// ---------------------------------------------------------------------------
__global__ void yolo_prep(const float* __restrict__ labels,
                          const float* __restrict__ anchors_all,
                          const int*   __restrict__ anchor_indices,
                          const int*   __restrict__ img_size_p,
                          char* __restrict__ blob)
{
  const int b = blockIdx.x;
  const int k = threadIdx.x;

  __shared__ int   s_n;
  __shared__ float s_anch[9][2];
  __shared__ int   s_aidx[N_A];

  if (threadIdx.x == 0) s_n = 0;
  if (threadIdx.x < 18) ((float*)s_anch)[threadIdx.x] = anchors_all[threadIdx.x];
  if (threadIdx.x < N_A) s_aidx[threadIdx.x] = anchor_indices[threadIdx.x];
  __syncthreads();

  const float img = (float)img_size_p[0];

  float l0 = 0.f, l1 = 0.f, l2 = 0.f, l3 = 0.f, l4 = 0.f;
  if (k < K_LAB) {
    const float* L = labels + ((size_t)b * K_LAB + k) * 5;
    l0 = L[0]; l1 = L[1]; l2 = L[2]; l3 = L[3]; l4 = L[4];
    if (l0 + l1 + l2 + l3 + l4 > 0.0f) atomicAdd(&s_n, 1);
  }
  __syncthreads();
  if (k >= K_LAB) return;

  const int  nlabel  = s_n;
  const bool valid_k = (k < nlabel);

  const int   tcls = (int)l0;
  const float tx = l1 * (float)NG;
  const float ty = l2 * (float)NG;
  const float tw = l3;
  const float th = l4;
  const int   ti = (int)tx;
  const int   tj = (int)ty;

  // argmax IoU of centered (0,0,tw,th) vs 9 centered anchors (first max wins)
  const float area_g = tw * th;
  int   best = 0;
  float best_iou = -1.0f;
  for (int n = 0; n < 9; ++n) {
    const float aw = s_anch[n][0] / img;
    const float ah = s_anch[n][1] / img;
    const float inter = fminf(tw, aw) * fminf(th, ah);
    const float uni   = fmaxf(area_g + aw * ah - inter, 1e-16f);
    const float iou   = inter / uni;
    if (iou > best_iou) { best_iou = iou; best = n; }
  }
  bool va = false;
  for (int t = 0; t < N_A; ++t) va = va || (best == s_aidx[t]);
  const bool valid  = valid_k && va;
  const int  best_n = best % N_A;

  char* B = blob + (size_t)b * BLOB_STRIDE;
  float4* gt_out  = (float4*)(B + BLOB_GT);
  float4* tgt_out = (float4*)(B + BLOB_TGT);
  float*  ts_out  = (float*)(B + BLOB_TS);
  int*    mt_out  = (int*)(B + BLOB_META);

  const float mk = valid_k ? 1.0f : 0.0f;
  gt_out[k] = make_float4(tx / (float)NG * mk, ty / (float)NG * mk,
                          tw * mk, th * mk);

  const float naw0 = s_anch[s_aidx[best_n]][0] / img;
  const float naw1 = s_anch[s_aidx[best_n]][1] / img;
  float4 tv;
  tv.x = tx - floorf(tx);
  tv.y = ty - floorf(ty);
  tv.z = __logf(tw / naw0 + 1e-16f);
  tv.w = __logf(th / naw1 + 1e-16f);
  tgt_out[k] = tv;
  ts_out[k]  = sqrtf(2.0f - tw * th);
  mt_out[k]  = (valid ? 1 : 0) | (best_n << 1) | ((ti & 0x7f) << 3) |
               ((tj & 0x7f) << 10) | ((tcls & 0x7f) << 17);
}

// ---------------------------------------------------------------------------
// Kernel 2: per-cell loss. grid = (ceil(nG*nG/256), N_A, nB), block = 256.
// ---------------------------------------------------------------------------
__global__ void __launch_bounds__(256)
yolo_main(const float* __restrict__ raw,
          const float* __restrict__ anchors_all,
          const int*   __restrict__ anchor_indices,
          const int*   __restrict__ img_size_p,
          const char*  __restrict__ blob,
          float* __restrict__ out)
{
  const int b    = blockIdx.z;
  const int a    = blockIdx.y;
  const int cell = blockIdx.x * blockDim.x + threadIdx.x;

  __shared__ alignas(16) char s_blob[BLOB_STRIDE];
  __shared__ float s_total;
  if (threadIdx.x == 0) s_total = 0.0f;

  // ---- stage the per-batch 2KB blob into LDS -----------------------------
  const char* gB = blob + (size_t)b * BLOB_STRIDE;
#if defined(__gfx1250__) &&                                                  \
    __has_builtin(__builtin_amdgcn_global_load_async_to_lds_b128)
  // CDNA5 async DMA path: 128 lanes x 16B, tracked with ASYNCcnt.
  // Builtin signature (per hipcc diagnostic): (v4i AS(1)*, v4i AS(3)*, Ii, Ii).
  if (threadIdx.x < BLOB_STRIDE / 16) {
    __builtin_amdgcn_global_load_async_to_lds_b128(
        (__attribute__((address_space(1))) v4i*)(gB + threadIdx.x * 16),
        (__attribute__((address_space(3))) v4i*)(s_blob + threadIdx.x * 16),
        /*offset=*/0, /*cpol=*/0);
  }
#if __has_builtin(__builtin_amdgcn_s_wait_asynccnt)
  __builtin_amdgcn_s_wait_asynccnt(0);
#else
  asm volatile("s_wait_asynccnt 0x0" ::: "memory");
#endif
#else
  if (threadIdx.x < BLOB_STRIDE / 16) {
    ((float4*)s_blob)[threadIdx.x] =
        ((const float4*)gB)[threadIdx.x];
  }
#endif
  __syncthreads();

  const float4* s_gt   = (const float4*)(s_blob + BLOB_GT);
  const float4* s_tgt  = (const float4*)(s_blob + BLOB_TGT);
  const float*  s_ts   = (const float*)(s_blob + BLOB_TS);
  const int*    s_meta = (const int*)(s_blob + BLOB_META);

  float loss = 0.0f;
  if (cell < NG * NG) {
    const int i = cell % NG;
    const int j = cell / NG;
    const float  img   = (float)img_size_p[0];
    const size_t plane = (size_t)NG * NG;
    const float* base  = raw + ((size_t)b * N_A + a) * NCH * plane + cell;

    // 5 always-needed channels (coalesced across i within the wave)
    const float rx = base[0 * plane];
    const float ry = base[1 * plane];
    const float rw = base[2 * plane];
    const float rh = base[3 * plane];
    const float rc = base[4 * plane];

    const float sx    = sigmoidf(rx);
    const float sy    = sigmoidf(ry);
    const float sconf = sigmoidf(rc);

    const float aw = anchors_all[anchor_indices[a] * 2 + 0] / img;
    const float ah = anchors_all[anchor_indices[a] * 2 + 1] / img;
    const float px = (sx + (float)i) / (float)NG;
    const float py = (sy + (float)j) / (float)NG;
    const float pw = fminf(__expf(rw) * aw, 1.0f);
    const float ph = fminf(__expf(rh) * ah, 1.0f);

    // max IoU of this predicted box vs all K gt boxes (LDS resident)
    const float phw = pw * 0.5f, phh = ph * 0.5f;
    const float parea = pw * ph;
    float maxiou = 0.0f;
    #pragma unroll 5
    for (int kk = 0; kk < K_LAB; ++kk) {
      const float4 g = s_gt[kk];
      const float ghw = g.z * 0.5f, ghh = g.w * 0.5f;
      float iw = fminf(px + phw, g.x + ghw) - fmaxf(px - phw, g.x - ghw);
      float ih = fminf(py + phh, g.y + ghh) - fmaxf(py - phh, g.y - ghh);
      iw = fmaxf(iw, 0.0f); ih = fmaxf(ih, 0.0f);
      const float inter = iw * ih;
      const float uni   = fmaxf(parea + g.z * g.w - inter, 1e-16f);
      maxiou = fmaxf(maxiou, inter / uni);
    }
    bool pen = !(maxiou > IGNORE_THRE);

    // target scatter emulation: last valid label mapping to (a, j, i) wins
    int hit = -1;
    for (int kk = 0; kk < K_LAB; ++kk) {
      const int m = s_meta[kk];
      if ((m & 1) && (((m >> 1) & 3) == a) &&
          (((m >> 3) & 0x7f) == i) && (((m >> 10) & 0x7f) == j))
        hit = kk;
    }

    float obj = 0.0f;
    if (hit >= 0) {
      obj = 1.0f;
      pen = true;
      const float4 tv = s_tgt[hit];
      const float  w2 = s_ts[hit] * s_ts[hit];
      loss += w2 * (bcef(sx, tv.x) + bcef(sy, tv.y));               // loss_xy
      const float dw = rw - tv.z, dh = rh - tv.w;
      loss += 0.5f * w2 * (dw * dw + dh * dh);                      // 0.5*loss_wh
      const int tc = (s_meta[hit] >> 17) & 0x7f;
      __builtin_prefetch(base + 5 * plane, 0, 1);                   // class chans
      for (int c = 0; c < N_CLS; ++c) {
        const float sc = sigmoidf(base[(size_t)(5 + c) * plane]);
        loss += bcef(sc, (c == tc) ? 1.0f : 0.0f);                  // loss_cls
      }
    }
    if (pen) loss += bcef(sconf, obj);                              // loss_obj
  }

  // ---- wave32 reduction via V_WMMA_F32_16X16X4_F32 -----------------------
  // A (16x4): lane L -> A[L%16][{0|2}] = loss, other slot 0. B = ones(4x16).
  // D[m][n] = p[m] + p[16+m]; lane 0 sums rows 0..7, lane 16 rows 8..15.
  // All 256 threads reach here (EXEC all ones; inactive cells carry loss=0).
  v2f Aop; Aop.x = loss; Aop.y = 0.0f;
  v2f Bop; Bop.x = 1.0f; Bop.y = 1.0f;
  v8f Cop = {0.f, 0.f, 0.f, 0.f, 0.f, 0.f, 0.f, 0.f};
  v8f D = __builtin_amdgcn_wmma_f32_16x16x4_f32(
      /*neg_a=*/false, Aop, /*neg_b=*/false, Bop,
      /*c_mod=*/(short)0, Cop, /*reuse_a=*/false, /*reuse_b=*/false);
  const float wsum = D[0] + D[1] + D[2] + D[3] + D[4] + D[5] + D[6] + D[7];

  const int lane = threadIdx.x & 31;   // wave32
  if (lane == 0 || lane == 16) atomicAdd(&s_total, wsum);
  __syncthreads();
  if (threadIdx.x == 0) atomicAdd(out, s_total);
}

// ---------------------------------------------------------------------------
// Host-side launch (graph-capture safe: only async ops on `stream`).
// Inputs: 0=raw f32, 1=labels f32, 2=anchors_all f32, 3=anchor_indices i32,
//         4=img_size i32 scalar. Output: 1 float (total loss).
// ---------------------------------------------------------------------------
extern "C" void kernel_launch(void* const* d_in, const int* in_sizes, int n_in,
                              void* d_out, int out_size, void* d_ws, size_t ws_size,
                              hipStream_t stream) {
  (void)in_sizes; (void)n_in; (void)out_size; (void)ws_size;

  const float* raw            = (const float*)d_in[0];
  const float* labels         = (const float*)d_in[1];
  const float* anchors_all    = (const float*)d_in[2];
  const int*   anchor_indices = (const int*)d_in[3];
  const int*   img_size       = (const int*)d_in[4];
  float*       out            = (float*)d_out;

  char* blob = (char*)d_ws;   // NB * BLOB_STRIDE = 32 KB of scratch

  (void)hipMemsetAsync(d_out, 0, sizeof(float), stream);

  yolo_prep<<<NB, 64, 0, stream>>>(labels, anchors_all, anchor_indices,
                                   img_size, blob);

  dim3 grid((NG * NG + 255) / 256, N_A, NB);
  yolo_main<<<grid, 256, 0, stream>>>(raw, anchors_all, anchor_indices,
                                      img_size, blob, out);
}